// Net_12652973654054
// MI455X (gfx1250) — compile-verified
//
#include <hip/hip_runtime.h>
#include <math.h>

#define F_IN 500
#define HID 256
#define NC 16
#define KSTEPS 10
#define KP1 (KSTEPS + 1)

typedef __attribute__((ext_vector_type(2))) float v2f;
typedef __attribute__((ext_vector_type(8))) float v8f;

// 508 words = 2032 B row pitch: 16B-aligned for async b128 writes, and
// 60*L mod 64 = 4*(15L mod 16) is a permutation -> 16 distinct banks per
// half-wave; the two K-halves differ by 2 mod 4 -> 32 distinct banks total.
#define XS_STRIDE 508
#define HS_STRIDE 257   // 257 mod 64 = 1 (odd) -> conflict-free column reads

// ---------------------------------------------------------------------------
// Fused  relu(x@W1+b1)@W2+b2  ->  z  (one 16-row tile per block, fp32 WMMA,
// x tile staged into LDS with CDNA5 async global->LDS B128 copies)
// ---------------------------------------------------------------------------
__global__ __launch_bounds__(256) void mlp_fused_kernel(
    const float* __restrict__ x, const float* __restrict__ W1,
    const float* __restrict__ b1, const float* __restrict__ W2,
    const float* __restrict__ b2, float* __restrict__ z, int N)
{
    __shared__ float xs[16 * XS_STRIDE];   // 16 x 500 input tile (padded)
    __shared__ float hs[16 * HS_STRIDE];   // 16 x 256 relu'd hidden tile

    const int row0 = blockIdx.x * 16;

    // Async-stage the 16x500 x-tile into LDS: 16 rows x 125 chunks of 16 B.
    for (int idx = threadIdx.x; idx < 16 * 125; idx += 256) {
        int r = idx / 125;
        int c = idx - r * 125;             // 16-byte chunk within row
        int gr = row0 + r;
        if (gr >= N) gr = N - 1;
        uint32_t lds_off = (uint32_t)(uintptr_t)(&xs[r * XS_STRIDE + c * 4]);
        const float* gp = x + (size_t)gr * F_IN + c * 4;
        asm volatile("global_load_async_to_lds_b128 %0, %1, off"
                     :: "v"(lds_off), "v"((unsigned long long)(uintptr_t)gp)
                     : "memory");
    }
    asm volatile("s_wait_asynccnt 0x0" ::: "memory");
    __syncthreads();

    const int wave = threadIdx.x >> 5;
    const int lane = threadIdx.x & 31;
    const int lh   = lane & 15;   // M (A) / N (B,C) index
    const int hi   = lane >> 4;   // K-half selector

    // GEMM1: each of 8 waves produces two 16x16 hidden tiles, K=500 in chunks of 4
    for (int t = 0; t < 2; ++t) {
        const int col0 = wave * 32 + t * 16;
        v8f acc = {};
        for (int k0 = 0; k0 < F_IN; k0 += 4) {
            const int kk = k0 + hi * 2;
            v2f a, b;
            a.x = xs[lh * XS_STRIDE + kk];
            a.y = xs[lh * XS_STRIDE + kk + 1];
            b.x = W1[kk * HID + col0 + lh];
            b.y = W1[(kk + 1) * HID + col0 + lh];
            acc = __builtin_amdgcn_wmma_f32_16x16x4_f32(
                false, a, false, b, (short)0, acc, false, false);
        }
        const float bias = b1[col0 + lh];
#pragma unroll
        for (int r = 0; r < 8; ++r) {
            float v = acc[r] + bias;
            hs[(r + hi * 8) * HS_STRIDE + col0 + lh] = v > 0.f ? v : 0.f;
        }
    }
    __syncthreads();

    // GEMM2: wave 0 multiplies the 16x256 hidden tile by W2 (256x16)
    if (wave == 0) {
        v8f acc = {};
        for (int k0 = 0; k0 < HID; k0 += 4) {
            const int kk = k0 + hi * 2;
            v2f a, b;
            a.x = hs[lh * HS_STRIDE + kk];
            a.y = hs[lh * HS_STRIDE + kk + 1];
            b.x = W2[kk * NC + lh];
            b.y = W2[(kk + 1) * NC + lh];
            acc = __builtin_amdgcn_wmma_f32_16x16x4_f32(
                false, a, false, b, (short)0, acc, false, false);
        }
        const float bias = b2[lh];
#pragma unroll
        for (int r = 0; r < 8; ++r) {
            const int gr = row0 + r + hi * 8;
            if (gr < N) z[gr * NC + lh] = acc[r] + bias;
        }
    }
}

// ---------------------------------------------------------------------------
// GCN normalization precompute
// ---------------------------------------------------------------------------
__global__ void deg_init_kernel(float* __restrict__ degf, int N) {
    int i = blockIdx.x * 256 + threadIdx.x;
    if (i < N) degf[i] = 1.0f;                 // self loop
}
__global__ void deg_count_kernel(const int* __restrict__ dst,
                                 float* __restrict__ degf, int E) {
    int e = blockIdx.x * 256 + threadIdx.x;
    if (e < E) atomicAdd(&degf[dst[e]], 1.0f);
}
__global__ void dinv_kernel(const float* __restrict__ degf,
                            float* __restrict__ dinv,
                            float* __restrict__ selfn, int N) {
    int i = blockIdx.x * 256 + threadIdx.x;
    if (i < N) {
        float d  = degf[i];
        dinv[i]  = rsqrtf(d);
        selfn[i] = 1.0f / d;                   // dinv^2 for the self loop
    }
}
__global__ void norm_kernel(const int* __restrict__ src, const int* __restrict__ dst,
                            const float* __restrict__ dinv,
                            float* __restrict__ nrm, int E) {
    int e = blockIdx.x * 256 + threadIdx.x;
    if (e < E) nrm[e] = dinv[src[e]] * dinv[dst[e]];
}

// ---------------------------------------------------------------------------
// Propagation: out = scale * (self-loop term) ; then edge scatter adds
// ---------------------------------------------------------------------------
__global__ void prop_init_kernel(float* __restrict__ out, const float* __restrict__ hin,
                                 const float* __restrict__ selfn, float scale, int N) {
    int i = blockIdx.x * 256 + threadIdx.x;
    if (i < N * NC) out[i] = scale * selfn[i >> 4] * hin[i];
}
__global__ void prop_scatter_kernel(float* __restrict__ out, const float* __restrict__ hin,
                                    const int* __restrict__ src, const int* __restrict__ dst,
                                    const float* __restrict__ nrm, float scale, int E) {
    int t = blockIdx.x * 256 + threadIdx.x;
    int e = t >> 4;                            // 16 lanes per edge (one per class)
    if (e < E) {
        int c = t & 15;
        int s = src[e], d = dst[e];
        atomicAdd(&out[d * NC + c], scale * nrm[e] * hin[s * NC + c]);
    }
}
__global__ void combine_kernel(float* __restrict__ pn, const float* __restrict__ t2,
                               const float* __restrict__ z, const float* __restrict__ pc,
                               float bk, int N) {
    int i = blockIdx.x * 256 + threadIdx.x;
    if (i < N * NC) pn[i] = 0.9f * t2[i] + 0.1f * z[i] + bk * pc[i];
}

// ---------------------------------------------------------------------------
// Final: sigmoid-scored mixture over K+1 predictions, then log_softmax
// 16 lanes per node; width-16 shuffle reductions.
// ---------------------------------------------------------------------------
__global__ __launch_bounds__(256) void final_kernel(
    const float* __restrict__ preds, const float* __restrict__ proj_w,
    const float* __restrict__ proj_b, float* __restrict__ out, int N)
{
    int tid  = blockIdx.x * 256 + threadIdx.x;
    int node = tid >> 4;
    int c    = tid & 15;
    if (node >= N) return;

    const float pw = proj_w[c];
    const float pb = proj_b[0];

    float vals[KP1];
#pragma unroll
    for (int k = 0; k < KP1; ++k)
        vals[k] = preds[(size_t)k * N * NC + node * NC + c];

    float acc = 0.f;
#pragma unroll
    for (int k = 0; k < KP1; ++k) {
        float d = vals[k] * pw;
        d += __shfl_xor(d, 1, 16);
        d += __shfl_xor(d, 2, 16);
        d += __shfl_xor(d, 4, 16);
        d += __shfl_xor(d, 8, 16);
        float s = 1.f / (1.f + expf(-(d + pb)));
        acc += s * vals[k];
    }

    float m = acc;
    m = fmaxf(m, __shfl_xor(m, 1, 16));
    m = fmaxf(m, __shfl_xor(m, 2, 16));
    m = fmaxf(m, __shfl_xor(m, 4, 16));
    m = fmaxf(m, __shfl_xor(m, 8, 16));
    float e  = expf(acc - m);
    float se = e;
    se += __shfl_xor(se, 1, 16);
    se += __shfl_xor(se, 2, 16);
    se += __shfl_xor(se, 4, 16);
    se += __shfl_xor(se, 8, 16);
    out[node * NC + c] = acc - m - logf(se);
}

// ---------------------------------------------------------------------------
extern "C" void kernel_launch(void* const* d_in, const int* in_sizes, int n_in,
                              void* d_out, int out_size, void* d_ws, size_t ws_size,
                              hipStream_t stream)
{
    const float* x  = (const float*)d_in[0];
    const int*   ei = (const int*)d_in[1];
    const float* W1 = (const float*)d_in[2];
    const float* b1 = (const float*)d_in[3];
    const float* W2 = (const float*)d_in[4];
    const float* b2 = (const float*)d_in[5];
    const float* pw = (const float*)d_in[6];
    const float* pb = (const float*)d_in[7];

    const int N = in_sizes[0] / F_IN;
    const int E = in_sizes[1] / 2;
    const int* src = ei;
    const int* dst = ei + E;

    // Workspace carve-up (floats): preds[11][N][16] | tmp1 | tmp2 | deg | dinv | selfn | norm[E]
    float* ws    = (float*)d_ws;
    float* preds = ws;
    float* tmp1  = preds + (size_t)KP1 * N * NC;
    float* tmp2  = tmp1 + (size_t)N * NC;
    float* degf  = tmp2 + (size_t)N * NC;
    float* dinv  = degf + N;
    float* selfn = dinv + N;
    float* nrm   = selfn + N;
    float* z     = preds;                       // preds[0] == z

    dim3 b256(256);
    const int gRows = (N + 15) / 16;
    const int gN    = (N + 255) / 256;
    const int gE    = (E + 255) / 256;
    const int gNC   = (N * NC + 255) / 256;
    const long long e16 = (long long)E * NC;
    const int gEC   = (int)((e16 + 255) / 256);

    // MLP (WMMA) -> preds[0]
    mlp_fused_kernel<<<gRows, b256, 0, stream>>>(x, W1, b1, W2, b2, z, N);

    // GCN norm precompute
    deg_init_kernel <<<gN, b256, 0, stream>>>(degf, N);
    deg_count_kernel<<<gE, b256, 0, stream>>>(dst, degf, E);
    dinv_kernel     <<<gN, b256, 0, stream>>>(degf, dinv, selfn, N);
    norm_kernel     <<<gE, b256, 0, stream>>>(src, dst, dinv, nrm, E);

    // K propagation steps: preds[k+1] = 0.9*P(P(0.5*preds[k])) + 0.1*z (+0.1*preds[k] if k>0)
    for (int k = 0; k < KSTEPS; ++k) {
        const float* pc = preds + (size_t)k * N * NC;
        float*       pn = preds + (size_t)(k + 1) * N * NC;
        const float  bk = (k > 0) ? 0.1f : 0.0f;
        prop_init_kernel   <<<gNC, b256, 0, stream>>>(tmp1, pc, selfn, 0.5f, N);
        prop_scatter_kernel<<<gEC, b256, 0, stream>>>(tmp1, pc, src, dst, nrm, 0.5f, E);
        prop_init_kernel   <<<gNC, b256, 0, stream>>>(tmp2, tmp1, selfn, 1.0f, N);
        prop_scatter_kernel<<<gEC, b256, 0, stream>>>(tmp2, tmp1, src, dst, nrm, 1.0f, E);
        combine_kernel     <<<gNC, b256, 0, stream>>>(pn, tmp2, z, pc, bk, N);
    }

    // Scored mixture + log_softmax -> d_out
    final_kernel<<<gNC, b256, 0, stream>>>(preds, pw, pb, (float*)d_out, N);
}